// Classifier_PatchCore_40922448396487
// MI455X (gfx1250) — compile-verified
//
#include <hip/hip_runtime.h>
#include <hip/hip_bf16.h>
#include <math.h>

// ---------------- problem constants ----------------
#define BATCH   8
#define PGRID   28
#define PP      (PGRID * PGRID)        // 784
#define NPATCH  (BATCH * PP)           // 6272
#define DIM     1536
#define MBANK   50000
#define KNN     9
#define IMG     224
#define IMGPIX  (IMG * IMG)            // 50176
#define OUTMAP  (BATCH * IMGPIX)       // 401408
#define KSIZE   33
#define KHALF   16

// distmin tiling: TileM = 32 (two 16-row WMMA tiles per workgroup),
// 196 row tiles; 25 bank chunks of 125 subtiles (16 cols each): 25*125*16 = 50000
#define TILEM        32
#define ROWTILES     (NPATCH / TILEM)  // 196
#define CHUNKS       25
#define SUBT_PER_CHUNK 125

typedef __attribute__((ext_vector_type(2))) float v2f;
typedef __attribute__((ext_vector_type(8))) float v8f;

// ---------------- helpers ----------------
__device__ inline float wave_sum(float v) {
    #pragma unroll
    for (int m = 16; m >= 1; m >>= 1) v += __shfl_xor(v, m, 32);
    return v;
}

// ---------------- row norms (||x||^2) ----------------
__global__ void __launch_bounds__(256)
row_norms_kernel(const float* __restrict__ x, float* __restrict__ out, int rows) {
    int wave = (blockIdx.x * blockDim.x + threadIdx.x) >> 5;
    int lane = threadIdx.x & 31;
    if (wave >= rows) return;
    const float* r = x + (size_t)wave * DIM;
    float s = 0.0f;
    for (int c = lane; c < DIM; c += 32) { float v = r[c]; s += v * v; }
    s = wave_sum(s);
    if (lane == 0) out[wave] = s;
}

// ---------------- init packed (dist,idx) mins ----------------
__global__ void __launch_bounds__(256)
init_best_kernel(unsigned long long* __restrict__ best) {
    int i = blockIdx.x * blockDim.x + threadIdx.x;
    if (i < NPATCH) best[i] = ~0ULL;
}

// ---------------- fused GEMM(dist^2) + min/argmin via WMMA f32 ----------------
// Each workgroup: 32 embedding rows staged in LDS (192 KB), one bank chunk.
// Each wave: one 16-wide bank subtile at a time, 2 M-tiles per B fragment load
// (one global_load_b64 feeds two v_wmma_f32_16x16x4_f32).
__global__ void __launch_bounds__(256)
distmin_wmma_kernel(const float* __restrict__ emb, const float* __restrict__ bank,
                    const float* __restrict__ xnorm, const float* __restrict__ ynorm,
                    unsigned long long* __restrict__ best) {
    extern __shared__ float As[];                 // TILEM x DIM fp32 tile of embedding
    const int m0 = blockIdx.x * TILEM;            // 32 embedding rows
    const int chunk = blockIdx.y;                 // bank chunk of 125 16-wide subtiles

    for (int i = threadIdx.x; i < TILEM * DIM; i += blockDim.x) {
        int rr = i / DIM, cc = i - rr * DIM;
        As[i] = emb[(size_t)(m0 + rr) * DIM + cc];
    }
    __syncthreads();

    const int waveId = threadIdx.x >> 5;
    const int lane   = threadIdx.x & 31;
    const int half   = lane >> 4;                 // 0 or 1
    const int r      = lane & 15;
    const int koff   = 2 * half;

    float xnv[2][8];
    #pragma unroll
    for (int t = 0; t < 2; ++t)
        #pragma unroll
        for (int v = 0; v < 8; ++v)
            xnv[t][v] = xnorm[m0 + t * 16 + v + 8 * half];

    float bestd[2][8];
    int   bestn[2][8];
    #pragma unroll
    for (int t = 0; t < 2; ++t)
        #pragma unroll
        for (int v = 0; v < 8; ++v) { bestd[t][v] = 3.4e38f; bestn[t][v] = 0x7fffffff; }

    const float* arow0 = As + r * DIM;            // M-tile 0, row r
    const float* arow1 = As + (16 + r) * DIM;     // M-tile 1, row r

    for (int st = waveId; st < SUBT_PER_CHUNK; st += 8) {   // uniform per wave
        const int n0 = chunk * (SUBT_PER_CHUNK * 16) + st * 16;
        const float* brow = bank + (size_t)(n0 + r) * DIM;  // this lane's B column

        v8f acc0 = {};
        v8f acc1 = {};
        #pragma unroll 4
        for (int k = 0; k < DIM; k += 4) {
            v2f b  = *(const v2f*)(brow + k + koff);        // one B load ...
            v2f a0 = *(const v2f*)(arow0 + k + koff);
            v2f a1 = *(const v2f*)(arow1 + k + koff);
            acc0 = __builtin_amdgcn_wmma_f32_16x16x4_f32(   // ... feeds two WMMAs
                false, a0, false, b, (short)0, acc0, false, false);
            acc1 = __builtin_amdgcn_wmma_f32_16x16x4_f32(
                false, a1, false, b, (short)0, acc1, false, false);
        }

        const float yn = ynorm[n0 + r];
        const int   n  = n0 + r;
        // n is strictly increasing per lane across subtiles -> strict '<' keeps
        // the first (smallest-index) minimum automatically.
        #pragma unroll
        for (int v = 0; v < 8; ++v) {
            float d0 = fmaxf(xnv[0][v] - 2.0f * acc0[v] + yn, 0.0f);
            if (d0 < bestd[0][v]) { bestd[0][v] = d0; bestn[0][v] = n; }
            float d1 = fmaxf(xnv[1][v] - 2.0f * acc1[v] + yn, 0.0f);
            if (d1 < bestd[1][v]) { bestd[1][v] = d1; bestn[1][v] = n; }
        }
    }

    // reduce over the 16 N-lanes within each half, then atomic combine
    #pragma unroll
    for (int t = 0; t < 2; ++t) {
        #pragma unroll
        for (int v = 0; v < 8; ++v) {
            float d = bestd[t][v]; int n = bestn[t][v];
            #pragma unroll
            for (int m = 8; m >= 1; m >>= 1) {
                float od = __shfl_xor(d, m, 32);
                int   on = __shfl_xor(n, m, 32);
                if (od < d || (od == d && on < n)) { d = od; n = on; }
            }
            if (r == 0) {
                unsigned long long key =
                    ((unsigned long long)__float_as_uint(d) << 32) | (unsigned int)n;
                atomicMin(&best[m0 + t * 16 + v + 8 * half], key);
            }
        }
    }
}

// ---------------- unpack best -> patch_scores (sqrt) + locations ----------------
__global__ void __launch_bounds__(256)
scores_kernel(const unsigned long long* __restrict__ best,
              float* __restrict__ ps, int* __restrict__ loc) {
    int i = blockIdx.x * blockDim.x + threadIdx.x;
    if (i < NPATCH) {
        unsigned long long k = best[i];
        ps[i]  = sqrtf(__uint_as_float((unsigned int)(k >> 32)));
        loc[i] = (int)(unsigned int)(k & 0xffffffffu);
    }
}

// ---------------- per-batch argmax of patch scores ----------------
__global__ void __launch_bounds__(256)
batch_argmax_kernel(const float* __restrict__ ps, const int* __restrict__ loc,
                    float* __restrict__ score, int* __restrict__ maxpatch,
                    int* __restrict__ nnidx) {
    int b = blockIdx.x;
    __shared__ float sv[256];
    __shared__ int   si[256];
    float bv = -1.0f; int bi = PP;
    for (int p = threadIdx.x; p < PP; p += 256) {
        float v = ps[b * PP + p];
        if (v > bv) { bv = v; bi = p; }
    }
    sv[threadIdx.x] = bv; si[threadIdx.x] = bi;
    __syncthreads();
    for (int s = 128; s > 0; s >>= 1) {
        if (threadIdx.x < s) {
            if (sv[threadIdx.x + s] > sv[threadIdx.x] ||
                (sv[threadIdx.x + s] == sv[threadIdx.x] && si[threadIdx.x + s] < si[threadIdx.x])) {
                sv[threadIdx.x] = sv[threadIdx.x + s];
                si[threadIdx.x] = si[threadIdx.x + s];
            }
        }
        __syncthreads();
    }
    if (threadIdx.x == 0) {
        score[b]    = sv[0];
        maxpatch[b] = si[0];
        nnidx[b]    = loc[b * PP + si[0]];
    }
}

// ---------------- d2: dist(nn_sample[b], bank) for all 8 batches, one bank pass ----------------
__global__ void __launch_bounds__(256)
nn_dist_kernel(const float* __restrict__ bank, const float* __restrict__ ynorm,
               const int* __restrict__ nnidx, float* __restrict__ d2buf, int nblocks) {
    int wave = (blockIdx.x * blockDim.x + threadIdx.x) >> 5;
    int lane = threadIdx.x & 31;
    const float* xr[BATCH];
    float xn[BATCH];
    #pragma unroll
    for (int b = 0; b < BATCH; ++b) {
        int idx = nnidx[b];
        xr[b] = bank + (size_t)idx * DIM;
        xn[b] = ynorm[idx];
    }
    int stride = nblocks * 8;
    for (int n = wave; n < MBANK; n += stride) {
        const float* yr = bank + (size_t)n * DIM;
        float s[BATCH] = {};
        for (int c = lane; c < DIM; c += 32) {
            float yv = yr[c];
            #pragma unroll
            for (int b = 0; b < BATCH; ++b) s[b] += xr[b][c] * yv;
        }
        #pragma unroll
        for (int b = 0; b < BATCH; ++b) {
            float t = wave_sum(s[b]);
            if (lane == 0)
                d2buf[(size_t)b * MBANK + n] = fmaxf(xn[b] - 2.0f * t + ynorm[n], 0.0f);
        }
    }
}

// ---------------- iterative 9-smallest selection per batch ----------------
__global__ void __launch_bounds__(256)
top9_kernel(const float* __restrict__ d2buf, int* __restrict__ supports) {
    int b = blockIdx.x;
    __shared__ int   sel[KNN];
    __shared__ float sv[256];
    __shared__ int   si[256];
    for (int k = 0; k < KNN; ++k) {
        float bv = 3.4e38f; int bi = MBANK;
        for (int n = threadIdx.x; n < MBANK; n += 256) {
            bool skip = false;
            for (int j = 0; j < k; ++j) if (sel[j] == n) skip = true;
            if (skip) continue;
            float v = d2buf[(size_t)b * MBANK + n];
            if (v < bv || (v == bv && n < bi)) { bv = v; bi = n; }
        }
        sv[threadIdx.x] = bv; si[threadIdx.x] = bi;
        __syncthreads();
        for (int s = 128; s > 0; s >>= 1) {
            if (threadIdx.x < s) {
                if (sv[threadIdx.x + s] < sv[threadIdx.x] ||
                    (sv[threadIdx.x + s] == sv[threadIdx.x] && si[threadIdx.x + s] < si[threadIdx.x])) {
                    sv[threadIdx.x] = sv[threadIdx.x + s];
                    si[threadIdx.x] = si[threadIdx.x + s];
                }
            }
            __syncthreads();
        }
        if (threadIdx.x == 0) { sel[k] = si[0]; supports[b * KNN + k] = si[0]; }
        __syncthreads();
    }
}

// ---------------- d3 + softmax reweight -> pred_score ----------------
__global__ void __launch_bounds__(256)
final_score_kernel(const float* __restrict__ emb, const float* __restrict__ bank,
                   const float* __restrict__ xnorm, const float* __restrict__ ynorm,
                   const int* __restrict__ supports, const int* __restrict__ maxpatch,
                   const float* __restrict__ score, float* __restrict__ out_pred) {
    int b = blockIdx.x;
    __shared__ float d3[KNN];
    int patch = maxpatch[b];
    const float* xr = emb + ((size_t)b * PP + patch) * DIM;
    float xn = xnorm[b * PP + patch];
    int wave = threadIdx.x >> 5, lane = threadIdx.x & 31;
    for (int k = wave; k < KNN; k += 8) {
        int idx = supports[b * KNN + k];
        const float* yr = bank + (size_t)idx * DIM;
        float s = 0.0f;
        for (int c = lane; c < DIM; c += 32) s += xr[c] * yr[c];
        s = wave_sum(s);
        if (lane == 0) d3[k] = sqrtf(fmaxf(xn - 2.0f * s + ynorm[idx], 0.0f));
    }
    __syncthreads();
    if (threadIdx.x == 0) {
        float mx = d3[0];
        for (int k = 1; k < KNN; ++k) mx = fmaxf(mx, d3[k]);
        float denom = 0.0f;
        for (int k = 0; k < KNN; ++k) denom += expf(d3[k] - mx);
        float w = 1.0f - expf(d3[0] - mx) / denom;
        out_pred[b] = w * score[b];
    }
}

// ---------------- bilinear 28 -> 224 (half-pixel, edge clamp) ----------------
__global__ void __launch_bounds__(256)
resize_kernel(const float* __restrict__ ps, float* __restrict__ amap) {
    int idx = blockIdx.x * blockDim.x + threadIdx.x;
    if (idx >= OUTMAP) return;
    int b = idx / IMGPIX;
    int rem = idx - b * IMGPIX;
    int oy = rem / IMG, ox = rem - oy * IMG;
    float sy = (oy + 0.5f) * ((float)PGRID / IMG) - 0.5f;
    float sx = (ox + 0.5f) * ((float)PGRID / IMG) - 0.5f;
    int y0 = (int)floorf(sy); float fy = sy - (float)y0;
    int x0 = (int)floorf(sx); float fx = sx - (float)x0;
    int y0c = min(max(y0, 0), PGRID - 1), y1c = min(max(y0 + 1, 0), PGRID - 1);
    int x0c = min(max(x0, 0), PGRID - 1), x1c = min(max(x0 + 1, 0), PGRID - 1);
    const float* p = ps + b * PP;
    float v = (1.0f - fy) * ((1.0f - fx) * p[y0c * PGRID + x0c] + fx * p[y0c * PGRID + x1c])
            +          fy * ((1.0f - fx) * p[y1c * PGRID + x0c] + fx * p[y1c * PGRID + x1c]);
    amap[idx] = v;
}

// ---------------- separable 33-tap gaussian, reflect padding ----------------
__device__ inline int reflect224(int t) {
    if (t < 0) t = -t;
    if (t > IMG - 1) t = 2 * (IMG - 1) - t;
    return t;
}

__global__ void __launch_bounds__(256)
blur_kernel(const float* __restrict__ in, float* __restrict__ out, int horizontal) {
    __shared__ float w[KSIZE];
    if (threadIdx.x < KSIZE) {
        float x = (float)threadIdx.x - (float)KHALF;
        w[threadIdx.x] = expf(-(x * x) / 32.0f);   // 2*sigma^2 = 32
    }
    __syncthreads();
    float wsum = 0.0f;
    #pragma unroll
    for (int i = 0; i < KSIZE; ++i) wsum += w[i];
    int idx = blockIdx.x * blockDim.x + threadIdx.x;
    if (idx >= OUTMAP) return;
    int b = idx / IMGPIX;
    int rem = idx - b * IMGPIX;
    int y = rem / IMG, x = rem - y * IMG;
    const float* img = in + (size_t)b * IMGPIX;
    float acc = 0.0f;
    if (horizontal) {
        for (int i = 0; i < KSIZE; ++i) acc += w[i] * img[y * IMG + reflect224(x + i - KHALF)];
    } else {
        for (int i = 0; i < KSIZE; ++i) acc += w[i] * img[reflect224(y + i - KHALF) * IMG + x];
    }
    out[idx] = acc / wsum;
}

// ---------------- host-side launch ----------------
extern "C" void kernel_launch(void* const* d_in, const int* in_sizes, int n_in,
                              void* d_out, int out_size, void* d_ws, size_t ws_size,
                              hipStream_t stream) {
    const float* emb  = (const float*)d_in[0];   // [6272, 1536]
    const float* bank = (const float*)d_in[1];   // [50000, 1536]
    float* out = (float*)d_out;                  // [401408 map] + [8 pred]

    // workspace carve-up
    char* w = (char*)d_ws;
    float* xnorm = (float*)w;                       w += (size_t)NPATCH * 4;      // 6272 f
    float* ynorm = (float*)w;                       w += (size_t)MBANK * 4;       // 50000 f
    unsigned long long* best = (unsigned long long*)w; w += (size_t)NPATCH * 8;   // 6272 u64 (8B aligned)
    float* ps    = (float*)w;                       w += (size_t)NPATCH * 4;
    int*   loc   = (int*)w;                         w += (size_t)NPATCH * 4;
    float* score = (float*)w;                       w += BATCH * 4;
    int*   maxpatch = (int*)w;                      w += BATCH * 4;
    int*   nnidx = (int*)w;                         w += BATCH * 4;
    int*   supports = (int*)w;                      w += BATCH * KNN * 4;
    w = (char*)(((size_t)w + 15) & ~(size_t)15);
    float* d2buf = (float*)w;                       w += (size_t)BATCH * MBANK * 4;
    float* amap  = (float*)w;                       w += (size_t)OUTMAP * 4;
    float* tmp   = (float*)w;                       w += (size_t)OUTMAP * 4;
    (void)in_sizes; (void)n_in; (void)out_size; (void)ws_size;

    // 1) row norms
    row_norms_kernel<<<(NPATCH + 7) / 8, 256, 0, stream>>>(emb, xnorm, NPATCH);
    row_norms_kernel<<<(MBANK + 7) / 8, 256, 0, stream>>>(bank, ynorm, MBANK);

    // 2) fused distance + min/argmin (WMMA f32 16x16x4, TileM=32, 192KB LDS)
    init_best_kernel<<<(NPATCH + 255) / 256, 256, 0, stream>>>(best);
    dim3 g(ROWTILES, CHUNKS);
    distmin_wmma_kernel<<<g, 256, TILEM * DIM * sizeof(float), stream>>>(emb, bank, xnorm, ynorm, best);

    // 3) per-patch scores + locations, per-batch argmax
    scores_kernel<<<(NPATCH + 255) / 256, 256, 0, stream>>>(best, ps, loc);
    batch_argmax_kernel<<<BATCH, 256, 0, stream>>>(ps, loc, score, maxpatch, nnidx);

    // 4) re-ranking: d2, top-9, softmax reweight
    nn_dist_kernel<<<196, 256, 0, stream>>>(bank, ynorm, nnidx, d2buf, 196);
    top9_kernel<<<BATCH, 256, 0, stream>>>(d2buf, supports);
    final_score_kernel<<<BATCH, 256, 0, stream>>>(emb, bank, xnorm, ynorm,
                                                  supports, maxpatch, score, out + OUTMAP);

    // 5) anomaly map: resize + gaussian blur (vertical then horizontal)
    resize_kernel<<<(OUTMAP + 255) / 256, 256, 0, stream>>>(ps, amap);
    blur_kernel<<<(OUTMAP + 255) / 256, 256, 0, stream>>>(amap, tmp, 0);
    blur_kernel<<<(OUTMAP + 255) / 256, 256, 0, stream>>>(tmp, out, 1);
}